// CentroidClassifier_72919954752028
// MI455X (gfx1250) — compile-verified
//
#include <hip/hip_runtime.h>
#include <math.h>

typedef __attribute__((ext_vector_type(2))) float v2f;
typedef __attribute__((ext_vector_type(8))) float v8f;

#define B_N 65536
#define C_N 1000
#define D_N 128
#define CP  1024   // columns padded to 1024
#define MT  32     // rows per workgroup

// ---- pre-pass: per-row squared norms ------------------------------------
__global__ __launch_bounds__(256)
void rowsq_kernel(const float* __restrict__ src, float* __restrict__ dst, int rows) {
  int r = blockIdx.x * blockDim.x + threadIdx.x;
  if (r >= rows) return;
  const float4* p = (const float4*)(src + (size_t)r * D_N);
  float s = 0.f;
#pragma unroll
  for (int i = 0; i < D_N / 4; ++i) {
    float4 v = p[i];
    s += v.x * v.x + v.y * v.y + v.z * v.z + v.w * v.w;
  }
  dst[r] = s;
}

// ---- fused GEMM (f32 WMMA) + corrections + softmax ----------------------
__global__ __launch_bounds__(256)
void centroid_fused_kernel(const float* __restrict__ x,
                           const float* __restrict__ centers,
                           const float* __restrict__ xsq,
                           const float* __restrict__ csq,
                           float* __restrict__ out_logits,
                           float* __restrict__ out_conf,
                           float* __restrict__ out_logconf) {
  __shared__ float lds[MT * CP];   // 128 KB logits stripe (WGP has 320 KB)

  const int tid   = threadIdx.x;
  const int wave  = tid >> 5;       // 0..7, each wave owns 128 columns
  const int lane  = tid & 31;
  const int half  = lane >> 4;      // upper/lower 16 lanes
  const int lrow  = lane & 15;
  const int m0    = blockIdx.x * MT;
  const int ncol0 = wave * 128;

  v8f acc0[8] = {};                 // rows m0 .. m0+15
  v8f acc1[8] = {};                 // rows m0+16 .. m0+31

  // A-fragment pointers (ISA 7.12.2, 32-bit 16x4 layout):
  // lanes 0-15 hold M=lane, K={k,k+1}; lanes 16-31 hold M=lane-16, K={k+2,k+3}.
  const float* aptr0 = x + (size_t)(m0 + lrow) * D_N + 2 * half;
  const float* aptr1 = aptr0 + 16 * D_N;

  // B-fragment base pointers (columns >= C_N clamped to 0; those tiles are
  // dead — masked to -inf below — and the loads hit L0).
  const float* bptr[8];
#pragma unroll
  for (int t = 0; t < 8; ++t) {
    int n  = ncol0 + t * 16 + lrow;
    int ns = (n < C_N) ? n : 0;
    bptr[t] = centers + (size_t)ns * D_N + 2 * half;
  }

  // Software-pipelined K loop: issue k+4 fragment loads (own registers, so
  // they clause) before consuming k's fragments in 16 WMMAs.
  v2f a0c = *(const v2f*)(aptr0);
  v2f a1c = *(const v2f*)(aptr1);
  v2f bc[8];
#pragma unroll
  for (int t = 0; t < 8; ++t) bc[t] = *(const v2f*)(bptr[t]);

  for (int kk = 0; kk < D_N - 4; kk += 4) {
    v2f a0n = *(const v2f*)(aptr0 + kk + 4);
    v2f a1n = *(const v2f*)(aptr1 + kk + 4);
    v2f bn[8];
#pragma unroll
    for (int t = 0; t < 8; ++t) bn[t] = *(const v2f*)(bptr[t] + kk + 4);

#pragma unroll
    for (int t = 0; t < 8; ++t)
      acc0[t] = __builtin_amdgcn_wmma_f32_16x16x4_f32(
          false, a0c, false, bc[t], (short)0, acc0[t], false, false);
#pragma unroll
    for (int t = 0; t < 8; ++t)
      acc1[t] = __builtin_amdgcn_wmma_f32_16x16x4_f32(
          false, a1c, false, bc[t], (short)0, acc1[t], false, false);

    a0c = a0n;
    a1c = a1n;
#pragma unroll
    for (int t = 0; t < 8; ++t) bc[t] = bn[t];
  }
  // epilogue: last k-step
#pragma unroll
  for (int t = 0; t < 8; ++t)
    acc0[t] = __builtin_amdgcn_wmma_f32_16x16x4_f32(
        false, a0c, false, bc[t], (short)0, acc0[t], false, false);
#pragma unroll
  for (int t = 0; t < 8; ++t)
    acc1[t] = __builtin_amdgcn_wmma_f32_16x16x4_f32(
        false, a1c, false, bc[t], (short)0, acc1[t], false, false);

  const float NEG_INF = -__builtin_inff();

  // logits = cross - 0.5*||x||^2 - 0.5*||c||^2 ; mask padded cols; stage to LDS.
  // C/D layout: VGPR i -> row (i + 8*half), column = lane-group column.
#pragma unroll
  for (int t = 0; t < 8; ++t) {
    int n = ncol0 + t * 16 + lrow;
    float cc = (n < C_N) ? csq[n] : 0.f;
#pragma unroll
    for (int i = 0; i < 8; ++i) {
      int m = i + 8 * half;
      float v0 = acc0[t][i] - 0.5f * xsq[m0 + m] - 0.5f * cc;
      float v1 = acc1[t][i] - 0.5f * xsq[m0 + m + 16] - 0.5f * cc;
      if (n >= C_N) { v0 = NEG_INF; v1 = NEG_INF; }
      lds[m * CP + n]        = v0;
      lds[(m + 16) * CP + n] = v1;
    }
  }
  __syncthreads();

  // Row softmax: each wave owns 4 whole rows; full-wave shuffle reductions,
  // 128B-contiguous stores.
  for (int rr = 0; rr < MT / 8; ++rr) {
    int r = wave * (MT / 8) + rr;

    float mx = NEG_INF;
#pragma unroll 4
    for (int q = 0; q < CP / 32; ++q) mx = fmaxf(mx, lds[r * CP + lane + q * 32]);
#pragma unroll
    for (int d = 16; d >= 1; d >>= 1) mx = fmaxf(mx, __shfl_xor(mx, d));

    float s = 0.f;
#pragma unroll 4
    for (int q = 0; q < CP / 32; ++q) s += __expf(lds[r * CP + lane + q * 32] - mx);
#pragma unroll
    for (int d = 16; d >= 1; d >>= 1) s += __shfl_xor(s, d);

    float logsum = mx + __logf(s);

    size_t base = (size_t)(m0 + r) * C_N;
    for (int q = 0; q < CP / 32; ++q) {
      int n = lane + q * 32;
      if (n < C_N) {
        float lo = lds[r * CP + n];
        float lc = lo - logsum;
        out_logits[base + n]  = lo;
        out_logconf[base + n] = lc;
        out_conf[base + n]    = __expf(lc);
      }
    }
  }
}

extern "C" void kernel_launch(void* const* d_in, const int* in_sizes, int n_in,
                              void* d_out, int out_size, void* d_ws, size_t ws_size,
                              hipStream_t stream) {
  const float* x       = (const float*)d_in[0];
  const float* centers = (const float*)d_in[1];

  float* xsq = (float*)d_ws;          // 65536 floats
  float* csq = xsq + B_N;             // 1000 floats

  float* out_logits  = (float*)d_out;
  float* out_conf    = out_logits + (size_t)B_N * C_N;
  float* out_logconf = out_conf   + (size_t)B_N * C_N;

  rowsq_kernel<<<B_N / 256, 256, 0, stream>>>(x, xsq, B_N);
  rowsq_kernel<<<(C_N + 255) / 256, 256, 0, stream>>>(centers, csq, C_N);

  centroid_fused_kernel<<<B_N / MT, 256, 0, stream>>>(
      x, centers, xsq, csq, out_logits, out_conf, out_logconf);
}